// FullVariationalLayer_52183852646871
// MI455X (gfx1250) — compile-verified
//
#include <hip/hip_runtime.h>

typedef __attribute__((ext_vector_type(2))) float v2f;
typedef __attribute__((ext_vector_type(8))) float v8f;

#define NQ 4
#define NL 2

// ---------------------------------------------------------------------------
// Kernel A: build the constant 16x16 complex circuit matrix M from weights and
// pack Re(M)/Im(M) into WMMA A-fragment order in d_ws.
//   GEMM is  D[k][m] = sum_j M[k][j] * V[m][j]   (A = M, B = V^T)
//   A-fragment (f32 16x4, v2f per lane): lane holds row k = lane&15,
//   K-cols j = 4*chunk + 2*(lane>=16) + {0,1}.
//   ws[0..255]   = Re fragments, layout frag[chunk][lane][v]
//   ws[256..511] = Im fragments
// ---------------------------------------------------------------------------
__global__ __launch_bounds__(256) void qsim_build_matrix(const float* __restrict__ w,
                                                         float* __restrict__ ws) {
    __shared__ float MrL[16][16];  // [col j][row k]
    __shared__ float MiL[16][16];
    const int t = threadIdx.x;
    if (t < 16) {
        // column t of M, applied gate-by-gate
        float cr[16], ci[16];
#pragma unroll
        for (int i = 0; i < 16; ++i) { cr[i] = (i == t) ? 1.0f : 0.0f; ci[i] = 0.0f; }
#pragma unroll
        for (int l = 0; l < NL; ++l) {
#pragma unroll
            for (int q = 0; q < NQ; ++q) {
                const int mask = 1 << (3 - q);
                // RX(theta): [[c, -i s], [-i s, c]]
                {
                    const float th = w[(l * NQ + q) * 3 + 0] * 0.5f;
                    const float c = cosf(th), s = sinf(th);
#pragma unroll
                    for (int i = 0; i < 16; ++i)
                        if (!(i & mask)) {
                            const int j = i | mask;
                            const float a0r = cr[i], a0i = ci[i], a1r = cr[j], a1i = ci[j];
                            cr[i] = c * a0r + s * a1i;  ci[i] = c * a0i - s * a1r;
                            cr[j] = s * a0i + c * a1r;  ci[j] = -s * a0r + c * a1i;
                        }
                }
                // RY(theta): [[c, -s], [s, c]]
                {
                    const float th = w[(l * NQ + q) * 3 + 1] * 0.5f;
                    const float c = cosf(th), s = sinf(th);
#pragma unroll
                    for (int i = 0; i < 16; ++i)
                        if (!(i & mask)) {
                            const int j = i | mask;
                            const float a0r = cr[i], a0i = ci[i], a1r = cr[j], a1i = ci[j];
                            cr[i] = c * a0r - s * a1r;  ci[i] = c * a0i - s * a1i;
                            cr[j] = s * a0r + c * a1r;  ci[j] = s * a0i + c * a1i;
                        }
                }
                // RZ(theta): diag(e^{-i th}, e^{+i th})
                {
                    const float th = w[(l * NQ + q) * 3 + 2] * 0.5f;
                    const float c = cosf(th), s = sinf(th);
#pragma unroll
                    for (int i = 0; i < 16; ++i)
                        if (!(i & mask)) {
                            const int j = i | mask;
                            const float a0r = cr[i], a0i = ci[i], a1r = cr[j], a1i = ci[j];
                            cr[i] = c * a0r + s * a0i;  ci[i] = c * a0i - s * a0r;
                            cr[j] = c * a1r - s * a1i;  ci[j] = c * a1i + s * a1r;
                        }
                }
            }
            // CNOT ring: CNOT(q, (q+1)%4) — pure row permutation
#pragma unroll
            for (int q = 0; q < NQ; ++q) {
                const int mc = 1 << (3 - q);
                const int mt = 1 << (3 - ((q + 1) & 3));
#pragma unroll
                for (int i = 0; i < 16; ++i)
                    if ((i & mc) && !(i & mt)) {
                        const int j = i | mt;
                        float tr = cr[i]; cr[i] = cr[j]; cr[j] = tr;
                        float ti = ci[i]; ci[i] = ci[j]; ci[j] = ti;
                    }
            }
        }
#pragma unroll
        for (int i = 0; i < 16; ++i) { MrL[t][i] = cr[i]; MiL[t][i] = ci[i]; }
    }
    __syncthreads();
    // pack into A-fragment order: t = chunk*64 + lane*2 + v
    // A[k][j] = M[k][j] = MrL[col=j][row=k], k = lane&15, j = 4c + 2h + v
    {
        const int c = t >> 6, rem = t & 63, lane = rem >> 1, v = rem & 1;
        const int h = lane >> 4, k = lane & 15;
        const int j = 4 * c + 2 * h + v;
        ws[t]       = MrL[j][k];
        ws[256 + t] = MiL[j][k];
    }
}

// ---------------------------------------------------------------------------
// Kernel B: one 16-sample tile per wave-iteration.
//   B-fragment of V (f32 4x16): lane holds column m = lane&15,
//   rows j = 4c + 2*(lane>=16) + v.  j bits: [q0 q1] = chunk, q2 = half,
//   q3 = v  ->  separable build.  8 WMMAs per tile (Re + Im chains).
//   D layout: lane holds sample m = lane&15, basis k = vgpr + 8*half.
//   Readout: in-register 8-pt Walsh over vgprs + one shfl_xor(16) per qubit;
//   half-0 lanes store float4 {z0,z1,z2,z3} coalesced.
// ---------------------------------------------------------------------------
__global__ __launch_bounds__(256) void qsim_circuit(const float* __restrict__ x,
                                                    const float* __restrict__ ws,
                                                    float* __restrict__ out,
                                                    int ntiles) {
    const int lane = threadIdx.x & 31;
    const int h = lane >> 4;
    const int m = lane & 15;
    const int wid = (blockIdx.x * blockDim.x + threadIdx.x) >> 5;
    const int nwaves = (gridDim.x * blockDim.x) >> 5;

    // preload constant A fragments (Re/Im), 4 K-chunks each
    v2f ar[4], ai[4];
#pragma unroll
    for (int c = 0; c < 4; ++c) {
        ar[c] = *(const v2f*)(ws + c * 64 + lane * 2);
        ai[c] = *(const v2f*)(ws + 256 + c * 64 + lane * 2);
    }

    for (int tile = wid; tile < ntiles; tile += nwaves) {
        // per-lane sample angles (B-layout column m)
        const float4 xa = ((const float4*)x)[tile * 16 + m];
        float s0, c0, s1, c1, s2, c2, s3, c3;
        __sincosf(xa.x * 0.5f, &s0, &c0);
        __sincosf(xa.y * 0.5f, &s1, &c1);
        __sincosf(xa.z * 0.5f, &s2, &c2);
        __sincosf(xa.w * 0.5f, &s3, &c3);
        float g01[4];
        g01[0] = c0 * c1; g01[1] = c0 * s1; g01[2] = s0 * c1; g01[3] = s0 * s1;
        const float e2 = h ? s2 : c2;
        const float b0 = e2 * c3, b1 = e2 * s3;  // v = 0 / 1 (qubit-3 bit)

        v8f accR = {0.f, 0.f, 0.f, 0.f, 0.f, 0.f, 0.f, 0.f};
        v8f accI = {0.f, 0.f, 0.f, 0.f, 0.f, 0.f, 0.f, 0.f};
#pragma unroll
        for (int c = 0; c < 4; ++c) {
            v2f b;
            b.x = g01[c] * b0;
            b.y = g01[c] * b1;
            accR = __builtin_amdgcn_wmma_f32_16x16x4_f32(false, ar[c], false, b,
                                                         (short)0, accR, false, false);
            accI = __builtin_amdgcn_wmma_f32_16x16x4_f32(false, ai[c], false, b,
                                                         (short)0, accI, false, false);
        }

        // probabilities: lane m holds p_k for k = v + 8h, v = 0..7
        v8f p;
#pragma unroll
        for (int e = 0; e < 8; ++e) p[e] = accR[e] * accR[e] + accI[e] * accI[e];

        // in-register 8-point Walsh over v: need W0 (sum), W4 (bit2), W2 (bit1), W1 (bit0)
        const float t0 = p[0] + p[1], t1 = p[0] - p[1];
        const float t2 = p[2] + p[3], t3 = p[2] - p[3];
        const float t4 = p[4] + p[5], t5 = p[4] - p[5];
        const float t6 = p[6] + p[7], t7 = p[6] - p[7];
        const float sa = t0 + t2, sb = t4 + t6;
        const float da = t0 - t2, db = t4 - t6;
        const float W0 = sa + sb;              // sum of all 8
        const float W4 = sa - sb;              // sign by bit2 of v  -> qubit 1
        const float W2 = da + db;              // sign by bit1 of v  -> qubit 2
        const float W1 = (t1 + t3) + (t5 + t7);// sign by bit0 of v  -> qubit 3

        // qubit 0: sign is the half bit (k bit3 = h)
        float z0 = h ? -W0 : W0;
        z0 += __shfl_xor(z0, 16, 32);
        const float z1 = W4 + __shfl_xor(W4, 16, 32);
        const float z2 = W2 + __shfl_xor(W2, 16, 32);
        const float z3 = W1 + __shfl_xor(W1, 16, 32);

        // both halves now hold identical results; half 0 stores coalesced b128
        if (h == 0) {
            float4 zo;
            zo.x = z0; zo.y = z1; zo.z = z2; zo.w = z3;
            ((float4*)out)[(size_t)tile * 16 + m] = zo;
        }
    }
}

// ---------------------------------------------------------------------------
extern "C" void kernel_launch(void* const* d_in, const int* in_sizes, int n_in,
                              void* d_out, int out_size, void* d_ws, size_t ws_size,
                              hipStream_t stream) {
    const float* x = (const float*)d_in[0];       // [B,4] f32
    const float* w = (const float*)d_in[1];       // [2,4,3] f32
    float* ws = (float*)d_ws;                     // 512 floats used
    float* out = (float*)d_out;                   // [B,4] f32
    const int batch = in_sizes[0] / 4;
    const int ntiles = batch / 16;                // B = 1<<20 -> 65536 tiles

    qsim_build_matrix<<<1, 256, 0, stream>>>(w, ws);
    qsim_circuit<<<2048, 256, 0, stream>>>(x, ws, out, ntiles);
}